// SimplifiedCTM_34892314313364
// MI455X (gfx1250) — compile-verified
//
#include <hip/hip_runtime.h>
#include <math.h>

typedef float v2f __attribute__((ext_vector_type(2)));
typedef float v8f __attribute__((ext_vector_type(8)));

#define BB 64
#define NN 1024
#define MM 32
#define TT 25
#define RR 512
#define SOo 512
#define ODo 10
#define TRACE_COLS (MM + TT)   // 57: sliding window, no shift copies

__device__ inline float wave_reduce_sum(float v) {
    for (int o = 16; o > 0; o >>= 1) v += __shfl_xor(v, o, 32);
    return v;
}
__device__ inline float sigmoidf(float x) { return 1.0f / (1.0f + expf(-x)); }

// ---------------------------------------------------------------------------
// Generic fp32 WMMA GEMM: Y[64 x Nout] = X[64 x K] @ W[Nout x K]^T + bias
// block = 128 threads (4 waves); wave w owns batch rows [16w,16w+16);
// blockIdx.x owns output cols [16*bx, 16*bx+16). K stepped by 4 per WMMA.
// A 16x4 layout (ISA 7.12.2): lanes 0-15 hold K=0(v0),K=1(v1); lanes 16-31 K=2,K=3.
// B 4x16 mirrored; C: VGPR j -> M = j + 8*(lane>>4), N = lane&15.
// ---------------------------------------------------------------------------
__global__ void wmma_gemm_xwT(const float* __restrict__ X, int ldx,
                              const float* __restrict__ W, int K,
                              const float* __restrict__ bias,
                              float* __restrict__ Y, int ldy) {
    const int lane = threadIdx.x & 31;
    const int w    = threadIdx.x >> 5;
    const int half = lane >> 4;
    const int r    = lane & 15;
    const int ncol = blockIdx.x * 16 + r;

    const float* xrow = X + (size_t)(w * 16 + r) * ldx;
    const float* wrow = W + (size_t)ncol * K;

    v8f acc = {0.f, 0.f, 0.f, 0.f, 0.f, 0.f, 0.f, 0.f};
#pragma unroll 4
    for (int k = 0; k < K; k += 4) {
        v2f a, b;
        a.x = xrow[k + 2 * half];
        a.y = xrow[k + 2 * half + 1];
        b.x = wrow[k + 2 * half];
        b.y = wrow[k + 2 * half + 1];
        acc = __builtin_amdgcn_wmma_f32_16x16x4_f32(false, a, false, b,
                                                    (short)0, acc, false, false);
    }
    const float bc = bias[ncol];
#pragma unroll
    for (int j = 0; j < 8; ++j)
        Y[(size_t)(w * 16 + j + 8 * half) * ldy + ncol] = acc[j] + bc;
}

// ---------------------------------------------------------------------------
// Row-wise LayerNorm (optionally + ReLU), in place. grid = 64 rows.
// ---------------------------------------------------------------------------
__global__ void ln_rows(float* __restrict__ Y, int ld, int C,
                        const float* __restrict__ gamma,
                        const float* __restrict__ beta, int do_relu) {
    const int b = blockIdx.x, tid = threadIdx.x;
    __shared__ float s1[256], s2[256];
    float sum = 0.f, sq = 0.f;
    for (int c = tid; c < C; c += 256) {
        float x = Y[(size_t)b * ld + c];
        sum += x; sq += x * x;
    }
    s1[tid] = sum; s2[tid] = sq; __syncthreads();
    for (int o = 128; o > 0; o >>= 1) {
        if (tid < o) { s1[tid] += s1[tid + o]; s2[tid] += s2[tid + o]; }
        __syncthreads();
    }
    const float mu   = s1[0] / C;
    const float var  = s2[0] / C - mu * mu;
    const float rstd = rsqrtf(var + 1e-5f);
    for (int c = tid; c < C; c += 256) {
        float x = Y[(size_t)b * ld + c];
        float v = (x - mu) * rstd * gamma[c] + beta[c];
        if (do_relu) v = fmaxf(v, 0.f);
        Y[(size_t)b * ld + c] = v;
    }
}

// ---------------------------------------------------------------------------
// syn epilogue: GLU over 64x2048 -> LN over N=1024 -> write trace col (32+t)
// ---------------------------------------------------------------------------
__global__ void glu_ln_to_trace(const float* __restrict__ S,
                                const float* __restrict__ gamma,
                                const float* __restrict__ beta,
                                float* __restrict__ trace, int t) {
    const int b = blockIdx.x, tid = threadIdx.x;
    __shared__ float s1[256], s2[256];
    float vs[4];
    float sum = 0.f, sq = 0.f;
#pragma unroll
    for (int i = 0; i < 4; ++i) {
        int n = tid + i * 256;
        float a = S[(size_t)b * 2048 + n];
        float g = S[(size_t)b * 2048 + 1024 + n];
        float v = a * sigmoidf(g);
        vs[i] = v; sum += v; sq += v * v;
    }
    s1[tid] = sum; s2[tid] = sq; __syncthreads();
    for (int o = 128; o > 0; o >>= 1) {
        if (tid < o) { s1[tid] += s1[tid + o]; s2[tid] += s2[tid + o]; }
        __syncthreads();
    }
    const float mu   = s1[0] / 1024.f;
    const float var  = s2[0] / 1024.f - mu * mu;
    const float rstd = rsqrtf(var + 1e-5f);
#pragma unroll
    for (int i = 0; i < 4; ++i) {
        int n = tid + i * 256;
        float o = (vs[i] - mu) * rstd * gamma[n] + beta[n];
        trace[((size_t)b * NN + n) * TRACE_COLS + MM + t] = o;
    }
}

// ---------------------------------------------------------------------------
// Fused per-neuron NLM: hh_pre[16x256] = trace_win[16x32] @ nlm_w1[n] (WMMA),
// then GLU + dot(nlm_w2[n]) reduction -> act written into PRE[:,512+n].
// grid = (1024 neurons, 4 batch tiles); block = 512 (16 waves, one per h-tile).
// ---------------------------------------------------------------------------
__global__ void nlm_step(const float* __restrict__ trace,
                         const float* __restrict__ w1,
                         const float* __restrict__ b1,
                         const float* __restrict__ w2,
                         const float* __restrict__ b2,
                         float* __restrict__ PRE, int t) {
    const int n  = blockIdx.x;
    const int mb = blockIdx.y;
    const int lane = threadIdx.x & 31;
    const int wv   = threadIdx.x >> 5;   // 0..15 -> h-tile
    const int half = lane >> 4;
    const int r    = lane & 15;

    __shared__ float shh[16 * 256];

    const float* Ab = trace + ((size_t)(mb * 16 + r) * NN + n) * TRACE_COLS + (t + 1);
    const float* Bb = w1 + (size_t)n * (MM * 256) + (wv * 16 + r);

    v8f acc = {0.f, 0.f, 0.f, 0.f, 0.f, 0.f, 0.f, 0.f};
#pragma unroll
    for (int k = 0; k < MM; k += 4) {
        v2f a, b;
        a.x = Ab[k + 2 * half];
        a.y = Ab[k + 2 * half + 1];
        b.x = Bb[(size_t)(k + 2 * half) * 256];
        b.y = Bb[(size_t)(k + 2 * half + 1) * 256];
        acc = __builtin_amdgcn_wmma_f32_16x16x4_f32(false, a, false, b,
                                                    (short)0, acc, false, false);
    }
#pragma unroll
    for (int j = 0; j < 8; ++j)
        shh[(j + 8 * half) * 256 + wv * 16 + r] = acc[j];
    __syncthreads();

    // wave wv reduces batch-row wv: GLU(a,g) * w2, 128 terms over 32 lanes
    const int row = wv;
    float part = 0.f;
#pragma unroll
    for (int h = lane; h < 128; h += 32) {
        float av = shh[row * 256 + h]       + b1[(size_t)n * 256 + h];
        float gv = shh[row * 256 + 128 + h] + b1[(size_t)n * 256 + 128 + h];
        part += av * sigmoidf(gv) * w2[(size_t)n * 128 + h];
    }
    part = wave_reduce_sum(part);
    if (lane == 0)
        PRE[(size_t)(mb * 16 + row) * 1536 + 512 + n] = part + b2[n];
}

// ---------------------------------------------------------------------------
// aO/bO update + syncO. act lives at PRE[:,512:]. 64*512 elements.
// ---------------------------------------------------------------------------
__global__ void sync_step(const float* __restrict__ PRE,
                          const int* __restrict__ il, const int* __restrict__ ir,
                          const float* __restrict__ decay,
                          float* __restrict__ aO, float* __restrict__ bO,
                          float* __restrict__ SY) {
    int i = blockIdx.x * blockDim.x + threadIdx.x;
    if (i >= BB * SOo) return;
    int b = i / SOo, j = i % SOo;
    float al = PRE[(size_t)b * 1536 + 512 + il[j]];
    float ar = PRE[(size_t)b * 1536 + 512 + ir[j]];
    float rO = expf(-fminf(fmaxf(decay[j], 0.f), 15.f));
    float a  = rO * aO[i] + al * ar;
    float bb = rO * bO[i] + 1.0f;
    aO[i] = a; bO[i] = bb;
    SY[i] = a * rsqrtf(bb + 1e-8f);
}

// ---------------------------------------------------------------------------
// Output head: pred = P2 @ op_w3^T + b3 (64x10), softmax entropy, scatter to out.
// predictions[b,od,t], certainties[b,c,t] (concat flat).
// ---------------------------------------------------------------------------
__global__ void head_kernel(const float* __restrict__ P2,
                            const float* __restrict__ w3,
                            const float* __restrict__ b3,
                            float* __restrict__ out, int t) {
    const int b = blockIdx.x, lane = threadIdx.x;
    __shared__ float sp[ODo];
    if (lane < ODo) {
        float s = b3[lane];
        for (int k = 0; k < 64; ++k) s += P2[(size_t)b * 64 + k] * w3[lane * 64 + k];
        sp[lane] = s;
    }
    __syncthreads();
    if (lane == 0) {
        float mx = sp[0];
        for (int i = 1; i < ODo; ++i) mx = fmaxf(mx, sp[i]);
        float e[ODo], sum = 0.f;
        for (int i = 0; i < ODo; ++i) { e[i] = expf(sp[i] - mx); sum += e[i]; }
        float ent = 0.f;
        for (int i = 0; i < ODo; ++i) {
            float p = e[i] / sum;
            ent -= p * logf(p + 1e-10f);
        }
        ent /= logf((float)ODo);
        for (int i = 0; i < ODo; ++i)
            out[(size_t)b * (ODo * TT) + i * TT + t] = sp[i];
        float* cert = out + BB * ODo * TT;
        cert[(size_t)b * (2 * TT) + 0 * TT + t] = ent;
        cert[(size_t)b * (2 * TT) + 1 * TT + t] = 1.0f - ent;
    }
}

// ---------------------------------------------------------------------------
// Init kernels
// ---------------------------------------------------------------------------
__global__ void init_state(const float* __restrict__ s0,
                           const int* __restrict__ il, const int* __restrict__ ir,
                           float* __restrict__ PRE,
                           float* __restrict__ aO, float* __restrict__ bO) {
    int i = blockIdx.x * blockDim.x + threadIdx.x;
    if (i < BB * NN) {
        int b = i >> 10, n = i & (NN - 1);
        PRE[(size_t)b * 1536 + 512 + n] = s0[n];
    }
    if (i < BB * SOo) {
        int j = i % SOo;
        aO[i] = s0[il[j]] * s0[ir[j]];
        bO[i] = 1.0f;
    }
}

__global__ void init_trace(const float* __restrict__ st, float* __restrict__ trace) {
    int i = blockIdx.x * blockDim.x + threadIdx.x;
    if (i >= BB * NN * MM) return;
    int m = i & (MM - 1);
    int n = (i >> 5) & (NN - 1);
    int b = i >> 15;
    trace[((size_t)b * NN + n) * TRACE_COLS + m] = st[n * MM + m];
}

// ---------------------------------------------------------------------------
extern "C" void kernel_launch(void* const* d_in, const int* in_sizes, int n_in,
                              void* d_out, int out_size, void* d_ws, size_t ws_size,
                              hipStream_t stream) {
    const float* x     = (const float*)d_in[0];
    const float* s0    = (const float*)d_in[1];
    const float* st    = (const float*)d_in[2];
    const float* ie_w1 = (const float*)d_in[3];
    const float* ie_b1 = (const float*)d_in[4];
    const float* ie_g1 = (const float*)d_in[5];
    const float* ie_n1 = (const float*)d_in[6];
    const float* ie_w2 = (const float*)d_in[7];
    const float* ie_b2 = (const float*)d_in[8];
    const float* ie_g2 = (const float*)d_in[9];
    const float* ie_n2 = (const float*)d_in[10];
    const float* ie_w3 = (const float*)d_in[11];
    const float* ie_b3 = (const float*)d_in[12];
    const float* kv_w  = (const float*)d_in[13];
    const float* kv_b  = (const float*)d_in[14];
    const float* kv_g  = (const float*)d_in[15];
    const float* kv_n  = (const float*)d_in[16];
    const float* inp_w = (const float*)d_in[19];
    const float* inp_b = (const float*)d_in[20];
    const float* op_w  = (const float*)d_in[21];   // out_proj_w
    const float* op_b  = (const float*)d_in[22];
    const float* syn_w = (const float*)d_in[23];
    const float* syn_b = (const float*)d_in[24];
    const float* syn_g = (const float*)d_in[25];
    const float* syn_n = (const float*)d_in[26];
    const float* nw1   = (const float*)d_in[27];
    const float* nb1   = (const float*)d_in[28];
    const float* nw2   = (const float*)d_in[29];
    const float* nb2   = (const float*)d_in[30];
    const float* dec_o = (const float*)d_in[32];
    const float* o_w1  = (const float*)d_in[33];
    const float* o_b1  = (const float*)d_in[34];
    const float* o_g1  = (const float*)d_in[35];
    const float* o_n1  = (const float*)d_in[36];
    const float* o_w2  = (const float*)d_in[37];
    const float* o_b2  = (const float*)d_in[38];
    const float* o_g2  = (const float*)d_in[39];
    const float* o_n2  = (const float*)d_in[40];
    const float* o_w3  = (const float*)d_in[41];
    const float* o_b3  = (const float*)d_in[42];
    const int*   iol   = (const int*)d_in[43];
    const int*   ior   = (const int*)d_in[44];

    float* ws = (float*)d_ws;
    float* PRE    = ws;                          // 64*1536
    float* SYNOUT = PRE    + 64 * 1536;          // 64*2048
    float* TRACE  = SYNOUT + 64 * 2048;          // 64*1024*57
    float* aO     = TRACE  + (size_t)64 * 1024 * TRACE_COLS;
    float* bO     = aO     + 64 * 512;
    float* SY     = bO     + 64 * 512;
    float* H1     = SY     + 64 * 512;           // 64*128
    float* H2     = H1     + 64 * 128;           // 64*64
    float* ENC    = H2     + 64 * 64;            // 64*512
    float* KV     = ENC    + 64 * 512;           // 64*512
    float* VT     = KV     + 64 * 512;           // 64*512
    float* P1     = VT     + 64 * 512;           // 64*256
    float* P2     = P1     + 64 * 256;           // 64*64

    float* out = (float*)d_out;

    // ---- encoder + attention preamble ----
    wmma_gemm_xwT<<<128 / 16, 128, 0, stream>>>(x, 784, ie_w1, 784, ie_b1, H1, 128);
    ln_rows<<<64, 256, 0, stream>>>(H1, 128, 128, ie_g1, ie_n1, 1);
    wmma_gemm_xwT<<<64 / 16, 128, 0, stream>>>(H1, 128, ie_w2, 128, ie_b2, H2, 64);
    ln_rows<<<64, 256, 0, stream>>>(H2, 64, 64, ie_g2, ie_n2, 1);
    wmma_gemm_xwT<<<512 / 16, 128, 0, stream>>>(H2, 64, ie_w3, 64, ie_b3, ENC, 512);
    wmma_gemm_xwT<<<512 / 16, 128, 0, stream>>>(ENC, 512, kv_w, 512, kv_b, KV, 512);
    ln_rows<<<64, 256, 0, stream>>>(KV, 512, 512, kv_g, kv_n, 0);
    // Wv = in_proj_w[2R:], bv = in_proj_b[2R:]
    wmma_gemm_xwT<<<512 / 16, 128, 0, stream>>>(KV, 512, inp_w + (size_t)2 * RR * RR, 512,
                                                inp_b + 2 * RR, VT, 512);
    // attn_out -> PRE[:, :512] (ldy = 1536)
    wmma_gemm_xwT<<<512 / 16, 128, 0, stream>>>(VT, 512, op_w, 512, op_b, PRE, 1536);

    init_state<<<(BB * NN + 255) / 256, 256, 0, stream>>>(s0, iol, ior, PRE, aO, bO);
    init_trace<<<(BB * NN * MM + 255) / 256, 256, 0, stream>>>(st, TRACE);

    // ---- recurrent loop ----
    for (int t = 0; t < TT; ++t) {
        // syn: PRE (64x1536) @ syn_w^T -> 64x2048
        wmma_gemm_xwT<<<2048 / 16, 128, 0, stream>>>(PRE, 1536, syn_w, 1536, syn_b,
                                                     SYNOUT, 2048);
        glu_ln_to_trace<<<64, 256, 0, stream>>>(SYNOUT, syn_g, syn_n, TRACE, t);
        nlm_step<<<dim3(NN, 4), 512, 0, stream>>>(TRACE, nw1, nb1, nw2, nb2, PRE, t);
        sync_step<<<(BB * SOo + 255) / 256, 256, 0, stream>>>(PRE, iol, ior, dec_o,
                                                              aO, bO, SY);
        wmma_gemm_xwT<<<256 / 16, 128, 0, stream>>>(SY, 512, o_w1, 512, o_b1, P1, 256);
        ln_rows<<<64, 256, 0, stream>>>(P1, 256, 256, o_g1, o_n1, 1);
        wmma_gemm_xwT<<<64 / 16, 128, 0, stream>>>(P1, 256, o_w2, 256, o_b2, P2, 64);
        ln_rows<<<64, 256, 0, stream>>>(P2, 64, 64, o_g2, o_n2, 1);
        head_kernel<<<64, 32, 0, stream>>>(P2, o_w3, o_b3, out, t);
    }
}